// MessagePassingLayer_12893491823090
// MI455X (gfx1250) — compile-verified
//
#include <hip/hip_runtime.h>
#include <stdint.h>

#define N_NODES 50000
#define N_EDGES 800000
#define D_FEAT  64
#define ROW_BYTES (D_FEAT * 4)      // 256 B per feature row
#define GROUPS_PER_BLOCK 16         // 256 threads / 16 lanes-per-edge
#define SCATTER_BLOCKS 1024
#define DEPTH 4                     // async pipeline stages (LDS buffers)
#define BUF_BYTES (GROUPS_PER_BLOCK * D_FEAT * 4)   // 4096 B per stage

// ---------------------------------------------------------------------------
// Native f32 global atomic add (global_atomic_add_f32, resolved in L2)
// ---------------------------------------------------------------------------
__device__ __forceinline__ void atom_add_f32(float* p, float v) {
    unsafeAtomicAdd(p, v);
}

// ---------------------------------------------------------------------------
// Kernel 1: zero output accumulator and degree scratch (buffers are poisoned)
// ---------------------------------------------------------------------------
__global__ void MPL_zero_kernel(float* __restrict__ out, int n_out4,
                                float* __restrict__ degs, int n_deg) {
    const int i      = blockIdx.x * blockDim.x + threadIdx.x;
    const int stride = gridDim.x * blockDim.x;
    float4 z = make_float4(0.f, 0.f, 0.f, 0.f);
    for (int k = i; k < n_out4; k += stride) ((float4*)out)[k] = z;
    for (int k = i; k < n_deg;  k += stride) degs[k] = 0.f;
}

// ---------------------------------------------------------------------------
// Kernel 2: out-degree and in-degree histograms (f32 atomics, counts exact;
// both 200 KB arrays are L2-resident so these resolve at L2 atomic units)
// ---------------------------------------------------------------------------
__global__ void MPL_degree_kernel(const int* __restrict__ src,
                                  const int* __restrict__ dst,
                                  float* __restrict__ deg_out,
                                  float* __restrict__ deg_in) {
    const int e = blockIdx.x * blockDim.x + threadIdx.x;
    if (e < N_EDGES) {
        atom_add_f32(&deg_out[src[e]], 1.0f);
        atom_add_f32(&deg_in [dst[e]], 1.0f);
    }
}

// ---------------------------------------------------------------------------
// Kernel 3: 4-deep async-pipelined gather -> scale -> scatter-add.
// 16 lanes cooperate on one edge (float4 per lane = 256 B row). Feature rows
// for iterations it..it+3 are kept in flight with global_load_async_to_lds_b128
// (GVS mode: SGPR feat base + per-lane 32-bit offset) into 4 rotating LDS
// buffers; s_wait_asynccnt (in-order completion) gates consumption. Trip
// count is wave-uniform and out-of-range lanes are clamped to a valid edge so
// EXEC stays full for every async issue and per-wave ASYNCcnt stays uniform.
// ---------------------------------------------------------------------------
__global__ __launch_bounds__(256)
void MPL_scatter_kernel(const float* __restrict__ feat,
                        const int*  __restrict__ src,
                        const int*  __restrict__ dst,
                        const float* __restrict__ deg_out,
                        float* __restrict__ out,
                        int n_groups_total) {
    __shared__ __align__(16) float smem[DEPTH][GROUPS_PER_BLOCK][D_FEAT];

    const int lg  = threadIdx.x >> 4;        // group within block (0..15)
    const int sub = threadIdx.x & 15;        // lane within group  (0..15)
    const int gid = blockIdx.x * GROUPS_PER_BLOCK + lg;
    const int G   = n_groups_total;
    const int niter = (N_EDGES + G - 1) / G; // uniform across all waves

    // wave-relative LDS byte offset of my 16B slot in stage 0 (low 32 bits of
    // the flat shared-aperture address are the LDS offset on gfx1250);
    // stage p lives at +p*BUF_BYTES.
    const uint32_t lds_base  = (uint32_t)(uintptr_t)(&smem[0][lg][sub * 4]);
    const uint32_t lane_byte = (uint32_t)(sub * 16);

    // ---- prologue: fill pipeline with loads for iterations 0..DEPTH-2 ----
#pragma unroll
    for (int p = 0; p < DEPTH - 1; ++p) {
        if (p < niter) {
            int e = gid + p * G;
            if (e >= N_EDGES) e = N_EDGES - 1;           // clamp, keep EXEC full
            const uint32_t off  = (uint32_t)src[e] * ROW_BYTES + lane_byte;
            const uint32_t ldsp = lds_base + (uint32_t)(p * BUF_BYTES);
            asm volatile("global_load_async_to_lds_b128 %0, %1, %2"
                         :: "v"(ldsp), "v"(off), "s"(feat) : "memory");
        }
    }

    for (int it = 0; it < niter; ++it) {
        const int e_cur = gid + it * G;

        // issue load for iteration it+DEPTH-1 into the slot just freed
        if (it + DEPTH - 1 < niter) {                    // uniform branch
            int en = e_cur + (DEPTH - 1) * G;
            if (en >= N_EDGES) en = N_EDGES - 1;
            const uint32_t off  = (uint32_t)src[en] * ROW_BYTES + lane_byte;
            const uint32_t ldsn =
                lds_base + (uint32_t)(((it + DEPTH - 1) & (DEPTH - 1)) * BUF_BYTES);
            asm volatile("global_load_async_to_lds_b128 %0, %1, %2"
                         :: "v"(ldsn), "v"(off), "s"(feat) : "memory");
        }

        // drain: oldest outstanding load (iteration it) must be complete.
        // outstanding-after-it = min(DEPTH-1, niter-1-it); waits are uniform.
        {
            const int rem = niter - 1 - it;
            if      (rem >= 3) asm volatile("s_wait_asynccnt 0x3" ::: "memory");
            else if (rem == 2) asm volatile("s_wait_asynccnt 0x2" ::: "memory");
            else if (rem == 1) asm volatile("s_wait_asynccnt 0x1" ::: "memory");
            else               asm volatile("s_wait_asynccnt 0x0" ::: "memory");
        }

        if (e_cur < N_EDGES) {
            const int s = src[e_cur];
            const int d = dst[e_cur];
            const float scale = __frsqrt_rn(fmaxf(deg_out[s], 1.0f));
            // each lane reads back exactly the 16B it async-loaded
            const float4 v =
                *(const float4*)&smem[it & (DEPTH - 1)][lg][sub * 4];
            float* o = out + (size_t)d * D_FEAT + sub * 4;
            atom_add_f32(o + 0, v.x * scale);
            atom_add_f32(o + 1, v.y * scale);
            atom_add_f32(o + 2, v.z * scale);
            atom_add_f32(o + 3, v.w * scale);
        }
    }
}

// ---------------------------------------------------------------------------
// Kernel 4: in-place scaling by deg_in^-0.5 (float4 granularity)
// ---------------------------------------------------------------------------
__global__ void MPL_scale_kernel(float* __restrict__ out,
                                 const float* __restrict__ deg_in) {
    const int i = blockIdx.x * blockDim.x + threadIdx.x;   // over N*16 float4s
    if (i < N_NODES * (D_FEAT / 4)) {
        const int n = i >> 4;                              // /(D_FEAT/4)
        const float sc = __frsqrt_rn(fmaxf(deg_in[n], 1.0f));
        float4 v = ((float4*)out)[i];
        v.x *= sc; v.y *= sc; v.z *= sc; v.w *= sc;
        ((float4*)out)[i] = v;
    }
}

// ---------------------------------------------------------------------------
extern "C" void kernel_launch(void* const* d_in, const int* in_sizes, int n_in,
                              void* d_out, int out_size, void* d_ws, size_t ws_size,
                              hipStream_t stream) {
    const float* feat = (const float*)d_in[0];
    const int*   src  = (const int*)  d_in[1];
    const int*   dst  = (const int*)  d_in[2];
    float* out = (float*)d_out;

    float* deg_out = (float*)d_ws;          // [N_NODES]
    float* deg_in  = deg_out + N_NODES;     // [N_NODES]

    MPL_zero_kernel<<<512, 256, 0, stream>>>(out, N_NODES * D_FEAT / 4,
                                             deg_out, 2 * N_NODES);

    MPL_degree_kernel<<<(N_EDGES + 255) / 256, 256, 0, stream>>>(src, dst,
                                                                 deg_out, deg_in);

    MPL_scatter_kernel<<<SCATTER_BLOCKS, 256, 0, stream>>>(
        feat, src, dst, deg_out, out, SCATTER_BLOCKS * GROUPS_PER_BLOCK);

    MPL_scale_kernel<<<(N_NODES * (D_FEAT / 4) + 255) / 256, 256, 0, stream>>>(
        out, deg_in);
}